// PointnetSAModule_65592740544725
// MI455X (gfx1250) — compile-verified
//
#include <hip/hip_runtime.h>
#include <math.h>

// ---------------- problem constants ----------------
#define BATCH    8
#define NPTS     8192
#define NPOINT   512
#define NSAMPLE  32
#define NITER    32
#define CFEAT    64
#define RADIUS2  0.25f
#define BN_EPS   1e-5f
#define ROWS     (BATCH * NPOINT * NSAMPLE)   // 131072
#define C0       67
#define C0P      68                           // padded to multiple of 4

typedef __attribute__((ext_vector_type(2))) float v2f;
typedef __attribute__((ext_vector_type(8))) float v8f;

// ============================================================
// k-means: mean + centroid init (one block per batch)
// ============================================================
__global__ __launch_bounds__(256)
void k_meaninit(const float* __restrict__ xyz, float* __restrict__ cent,
                float* __restrict__ meanb) {
    int b = blockIdx.x, tid = threadIdx.x;
    __shared__ float sx[256], sy[256], sz[256];
    __shared__ float mn[3];
    float ax = 0.f, ay = 0.f, az = 0.f;
    for (int m = tid; m < NPTS; m += 256) {
        const float* p = xyz + ((long)b * NPTS + m) * 3;
        ax += p[0]; ay += p[1]; az += p[2];
    }
    sx[tid] = ax; sy[tid] = ay; sz[tid] = az;
    __syncthreads();
    for (int s = 128; s > 0; s >>= 1) {
        if (tid < s) { sx[tid] += sx[tid+s]; sy[tid] += sy[tid+s]; sz[tid] += sz[tid+s]; }
        __syncthreads();
    }
    if (tid == 0) {
        mn[0] = sx[0] / NPTS; mn[1] = sy[0] / NPTS; mn[2] = sz[0] / NPTS;
        meanb[b*3+0] = mn[0]; meanb[b*3+1] = mn[1]; meanb[b*3+2] = mn[2];
    }
    __syncthreads();
    for (int k = tid; k < NPOINT; k += 256) {
        float* c = cent + ((long)b * NPOINT + k) * 3;
        if (k == 0) { c[0] = mn[0]; c[1] = mn[1]; c[2] = mn[2]; }
        else {
            const float* p = xyz + ((long)b * NPTS + (k - 1)) * 3;
            c[0] = p[0]; c[1] = p[1]; c[2] = p[2];
        }
    }
}

__global__ void k_zero(float* __restrict__ p, int n) {
    int i = blockIdx.x * blockDim.x + threadIdx.x;
    if (i < n) p[i] = 0.f;
}

// assign: one thread per point, centroids staged in LDS
__global__ __launch_bounds__(256)
void k_assign(const float* __restrict__ xyz, const float* __restrict__ meanb,
              const float* __restrict__ cent, float* __restrict__ sums,
              float* __restrict__ cnts) {
    int b = blockIdx.y;
    int m = blockIdx.x * 256 + threadIdx.x;   // m in [0, 8192] => 8193 points
    __shared__ float cs[NPOINT * 3];
    const float* cb = cent + (long)b * NPOINT * 3;
    for (int i = threadIdx.x; i < NPOINT * 3; i += 256) cs[i] = cb[i];
    __syncthreads();
    if (m >= NPTS + 1) return;
    float px, py, pz;
    if (m == 0) { px = meanb[b*3]; py = meanb[b*3+1]; pz = meanb[b*3+2]; }
    else {
        const float* p = xyz + ((long)b * NPTS + (m - 1)) * 3;
        px = p[0]; py = p[1]; pz = p[2];
    }
    int best = 0; float bd = 3.4e38f;
    for (int k = 0; k < NPOINT; ++k) {
        float dx = px - cs[k*3], dy = py - cs[k*3+1], dz = pz - cs[k*3+2];
        float d = dx*dx + dy*dy + dz*dz;
        if (d < bd) { bd = d; best = k; }
    }
    long base = ((long)b * NPOINT + best);
    atomicAdd(&sums[base*3+0], px);
    atomicAdd(&sums[base*3+1], py);
    atomicAdd(&sums[base*3+2], pz);
    atomicAdd(&cnts[base], 1.f);
}

__global__ void k_update(float* __restrict__ cent, const float* __restrict__ sums,
                         const float* __restrict__ cnts, float* __restrict__ outp) {
    int i = blockIdx.x * blockDim.x + threadIdx.x;
    if (i >= BATCH * NPOINT) return;
    float c = fmaxf(cnts[i], 1.f);
    float x = sums[i*3+0] / c, y = sums[i*3+1] / c, z = sums[i*3+2] / c;
    cent[i*3+0] = x; cent[i*3+1] = y; cent[i*3+2] = z;
    if (outp) { outp[i*3+0] = x; outp[i*3+1] = y; outp[i*3+2] = z; }
}

// ============================================================
// ball query (wave32 ordered compaction via ballot) + gather
// builds X0 rows: [relx, rely, relz, feat0..63, 0pad]
// ============================================================
__global__ __launch_bounds__(128)
void k_ballgather(const float* __restrict__ xyz, const float* __restrict__ feat,
                  const float* __restrict__ cent, float* __restrict__ X0) {
    int tid = threadIdx.x;
    int wv = tid >> 5, l = tid & 31;
    int wid = blockIdx.x * 4 + wv;            // 0..4095  (b*512 + s)
    int b = wid >> 9;
    __shared__ int idxs[4][NSAMPLE];
    const float* c = cent + (long)wid * 3;
    float cx = c[0], cy = c[1], cz = c[2];
    int cnt = 0;
    for (int base = 0; base < NPTS && cnt < NSAMPLE; base += 32) {
        int n = base + l;
        const float* p = xyz + ((long)b * NPTS + n) * 3;
        float dx = p[0]-cx, dy = p[1]-cy, dz = p[2]-cz;
        bool hit = (dx*dx + dy*dy + dz*dz) < RADIUS2;
        unsigned mask = (unsigned)__ballot(hit);          // wave32: low 32 bits
        int pos = cnt + __popc(mask & ((1u << l) - 1u));
        if (hit && pos < NSAMPLE) idxs[wv][pos] = n;
        cnt += __popc(mask);
    }
    if (cnt == 0) { if (l == 0) idxs[wv][0] = 0; cnt = 1; }
    int first = idxs[wv][0];
    int myidx = (l < cnt) ? idxs[wv][l] : first;

    long row = (long)wid * NSAMPLE + l;
    float* xr = X0 + row * C0P;
    const float* p = xyz + ((long)b * NPTS + myidx) * 3;
    xr[0] = p[0] - cx; xr[1] = p[1] - cy; xr[2] = p[2] - cz;
    const float* fb = feat + (long)b * CFEAT * NPTS + myidx;
    #pragma unroll 8
    for (int ch = 0; ch < CFEAT; ++ch) xr[3 + ch] = fb[(long)ch * NPTS];
    xr[67] = 0.f;
}

// ============================================================
// fused GEMM (+ optional BN+ReLU on input) with fp32 WMMA
//   Y(R x NCOL) = act(X) * W^T ; W is (NCOL x CIN) row-major
//   epilogue accumulates per-channel sum / sumsq for next BN
// A frag (16x4 f32): lane<16 -> K=0,1 ; lane>=16 -> K=2,3
// D (16x16 f32): vgpr r, lane l -> M = r + 8*(l>>4), N = l&15
// ============================================================
template<int CIN, int CINP, int NCOL, bool BNIN>
__global__ __launch_bounds__(256)
void k_gemm(const float* __restrict__ X, const float* __restrict__ W,
            const float* __restrict__ bna, const float* __restrict__ bnc,
            float* __restrict__ Y, float* __restrict__ gsum, float* __restrict__ gsq) {
    constexpr int WPR = NCOL / 64;                 // waves per row tile (1 or 2)
    constexpr int RPB = (8 / WPR) * 16;            // rows per block (128 or 64)
    __shared__ float Ws[NCOL * CINP];
    __shared__ float bsum[NCOL], bsq[NCOL];
    __shared__ float bnaS[BNIN ? CIN : 1], bncS[BNIN ? CIN : 1];

    int tid = threadIdx.x;
    for (int i = tid; i < NCOL * CINP; i += 256) {
        int n = i / CINP, cc = i % CINP;
        Ws[i] = (cc < CIN) ? W[n * CIN + cc] : 0.f;
    }
    for (int i = tid; i < NCOL; i += 256) { bsum[i] = 0.f; bsq[i] = 0.f; }
    if constexpr (BNIN) {
        for (int i = tid; i < CIN; i += 256) { bnaS[i] = bna[i]; bncS[i] = bnc[i]; }
    }
    __syncthreads();

    int w = tid >> 5, l = tid & 31;
    int ln = l & 15, hl = l >> 4;
    int rowTile = w / WPR, colBase = (w % WPR) * 64;
    long rowBase = (long)blockIdx.x * RPB + rowTile * 16;
    long row = rowBase + ln;

    v8f acc[4] = {};
    const float* xrow = X + row * CINP;

    for (int kc = 0; kc < CINP / 4; ++kc) {
        int kk = kc * 4 + 2 * hl;
        v2f a;
        if constexpr (BNIN) {
            float x0 = xrow[kk], x1 = xrow[kk + 1];
            a.x = fmaxf(bnaS[kk]     * x0 + bncS[kk],     0.f);
            a.y = fmaxf(bnaS[kk + 1] * x1 + bncS[kk + 1], 0.f);
        } else {
            a = *reinterpret_cast<const v2f*>(xrow + kk);   // 8B aligned
        }
        #pragma unroll
        for (int t = 0; t < 4; ++t) {
            int n = colBase + t * 16 + ln;
            v2f bf;
            bf.x = Ws[n * CINP + kk];
            bf.y = Ws[n * CINP + kk + 1];
            acc[t] = __builtin_amdgcn_wmma_f32_16x16x4_f32(
                false, a, false, bf, (short)0, acc[t], false, false);
        }
    }

    #pragma unroll
    for (int t = 0; t < 4; ++t) {
        int col = colBase + t * 16 + ln;
        float ls = 0.f, lq = 0.f;
        #pragma unroll
        for (int r = 0; r < 8; ++r) {
            long rr = rowBase + r + 8 * hl;
            float v = acc[t][r];
            Y[rr * NCOL + col] = v;
            ls += v; lq += v * v;
        }
        ls += __shfl_xor(ls, 16, 32);
        lq += __shfl_xor(lq, 16, 32);
        if (hl == 0) { atomicAdd(&bsum[col], ls); atomicAdd(&bsq[col], lq); }
    }
    __syncthreads();
    for (int i = tid; i < NCOL; i += 256) {
        atomicAdd(&gsum[i], bsum[i]);
        atomicAdd(&gsq[i],  bsq[i]);
    }
}

__global__ void k_bnfin(const float* __restrict__ gsum, const float* __restrict__ gsq,
                        const float* __restrict__ g, const float* __restrict__ b,
                        float* __restrict__ a, float* __restrict__ c, int nch, float invR) {
    int i = blockIdx.x * blockDim.x + threadIdx.x;
    if (i >= nch) return;
    float m = gsum[i] * invR;
    float v = gsq[i] * invR - m * m;
    float s = g[i] * rsqrtf(v + BN_EPS);
    a[i] = s; c[i] = b[i] - m * s;
}

// BN3 + ReLU + max over nsample, transposed write (B,128,S)
__global__ __launch_bounds__(128)
void k_maxpool(const float* __restrict__ Y3, const float* __restrict__ a3,
               const float* __restrict__ c3, float* __restrict__ out) {
    int bs = blockIdx.x;            // b*512 + s
    int o = threadIdx.x;            // channel
    int b = bs >> 9, s = bs & 511;
    const float* yb = Y3 + (long)bs * NSAMPLE * 128 + o;
    float aa = a3[o], cc = c3[o];
    float m = 0.f;                   // ReLU floor
    #pragma unroll 4
    for (int n = 0; n < NSAMPLE; ++n)
        m = fmaxf(m, fmaxf(aa * yb[n * 128] + cc, 0.f));
    out[((long)b * 128 + o) * 512 + s] = m;
}

// ============================================================
extern "C" void kernel_launch(void* const* d_in, const int* in_sizes, int n_in,
                              void* d_out, int out_size, void* d_ws, size_t ws_size,
                              hipStream_t stream) {
    const float* xyz  = (const float*)d_in[0];
    const float* feat = (const float*)d_in[1];
    const float* w1 = (const float*)d_in[2];
    const float* g1 = (const float*)d_in[3];
    const float* b1 = (const float*)d_in[4];
    const float* w2 = (const float*)d_in[5];
    const float* g2 = (const float*)d_in[6];
    const float* b2 = (const float*)d_in[7];
    const float* w3 = (const float*)d_in[8];
    const float* g3 = (const float*)d_in[9];
    const float* b3 = (const float*)d_in[10];
    float* out = (float*)d_out;
    float* ws  = (float*)d_ws;

    // workspace layout (floats); total ~34.1M floats (~137 MB)
    float* cent  = ws;                         // 12288
    float* ksum  = cent + 12288;               // 12288
    float* kcnt  = ksum + 12288;               // 4096
    float* meanb = kcnt + 4096;                // 32
    float* bn    = meanb + 32;                 // 1024
    float* s1 = bn,        *q1 = bn + 64,  *a1 = bn + 128, *c1 = bn + 192;
    float* s2 = bn + 256,  *q2 = bn + 320, *a2 = bn + 384, *c2 = bn + 448;
    float* s3 = bn + 512,  *q3 = bn + 640, *a3 = bn + 768, *c3 = bn + 896;
    float* X0 = bn + 1024;                     // ROWS*68
    float* Y1 = X0 + (long)ROWS * C0P;         // ROWS*64
    float* Y3 = Y1 + (long)ROWS * 64;          // ROWS*128
    float* Y2 = X0;                            // reuse (X0 dead after GEMM1)

    // --- k-means (32 Lloyd iterations; final one writes new_xyz to d_out) ---
    k_meaninit<<<BATCH, 256, 0, stream>>>(xyz, cent, meanb);
    for (int it = 0; it < NITER; ++it) {
        k_zero<<<64, 256, 0, stream>>>(ksum, 12288 + 4096);
        k_assign<<<dim3(33, BATCH), 256, 0, stream>>>(xyz, meanb, cent, ksum, kcnt);
        k_update<<<16, 256, 0, stream>>>(cent, ksum, kcnt,
                                         (it == NITER - 1) ? out : nullptr);
    }

    // --- ball query + gather -> X0 ---
    k_ballgather<<<1024, 128, 0, stream>>>(xyz, feat, cent, X0);

    // --- MLP: three WMMA GEMMs with fused BN-stat epilogues ---
    const float invR = 1.0f / (float)ROWS;
    k_zero<<<4, 256, 0, stream>>>(bn, 1024);
    k_gemm<C0, C0P, 64, false><<<ROWS / 128, 256, 0, stream>>>(X0, w1, nullptr, nullptr, Y1, s1, q1);
    k_bnfin<<<1, 64, 0, stream>>>(s1, q1, g1, b1, a1, c1, 64, invR);
    k_gemm<64, 64, 64, true><<<ROWS / 128, 256, 0, stream>>>(Y1, w2, a1, c1, Y2, s2, q2);
    k_bnfin<<<1, 64, 0, stream>>>(s2, q2, g2, b2, a2, c2, 64, invR);
    k_gemm<64, 64, 128, true><<<ROWS / 64, 256, 0, stream>>>(Y2, w3, a2, c2, Y3, s3, q3);
    k_bnfin<<<1, 128, 0, stream>>>(s3, q3, g3, b3, a3, c3, 128, invR);

    // --- BN3 + ReLU + max over samples -> d_out[12288:] as (B,128,S) ---
    k_maxpool<<<BATCH * NPOINT, 128, 0, stream>>>(Y3, a3, c3, out + 12288);
}